// StridedAttention_12120397709566
// MI455X (gfx1250) — compile-verified
//
#include <hip/hip_runtime.h>

typedef __attribute__((ext_vector_type(2))) float v2f;
typedef __attribute__((ext_vector_type(8))) float v8f;

#define S_LEN   2048
#define NHEAD   16
#define DH      64
#define KSTRIDE 32
#define WIN     16
#define NTILES  7      // 3 local (48 positions) + 4 strided (64 positions)
#define NCOLS   112    // 48 local + 64 strided score columns
#define ROWPAD  68     // floats per LDS row (bank-conflict-free)
#define SCPAD   120    // floats per score row

__global__ __launch_bounds__(32)
void strided_attn_wmma_f32(const float* __restrict__ q,
                           const float* __restrict__ k,
                           const float* __restrict__ v,
                           float* __restrict__ out)
{
    __shared__ float q_lds[16 * ROWPAD];
    __shared__ float k_lds[16 * ROWPAD];
    __shared__ float v_lds[16 * ROWPAD];
    __shared__ float sc_lds[16 * SCPAD];

    const int lane  = threadIdx.x;     // 0..31 (wave32)
    const int m     = lane & 15;       // row/col index within 16
    const int hi    = lane >> 4;       // half-wave select (K-pair select for WMMA ops)
    const int qb    = blockIdx.x;      // 0..127  query block
    const int h     = blockIdx.y;      // 0..15   head
    const int qbase = qb * 16;
    const long hbase = (long)h * S_LEN * DH;

    // ---- stage Q tile (16 x 64): one row per iteration, 32 lanes x float2 ----
    for (int r = 0; r < 16; ++r) {
        const float* src = q + hbase + (long)(qbase + r) * DH + 2 * lane;
        v2f t; t.x = src[0]; t.y = src[1];
        *(v2f*)&q_lds[r * ROWPAD + 2 * lane] = t;
    }
    __syncthreads();

    // ================= score phase: S = (Q K^T) * scale, masked =================
    for (int t = 0; t < NTILES; ++t) {
        // stage K tile rows into LDS (row-major [key][d])
        for (int r = 0; r < 16; ++r) {
            int pos;
            if (t < 3) {
                pos = qbase - WIN + t * 16 + r;           // local span
                pos = pos < 0 ? 0 : (pos > S_LEN - 1 ? S_LEN - 1 : pos);
            } else {
                pos = ((t - 3) * 16 + r) * KSTRIDE;       // strided keys
            }
            const float* src = k + hbase + (long)pos * DH + 2 * lane;
            v2f tv; tv.x = src[0]; tv.y = src[1];
            *(v2f*)&k_lds[r * ROWPAD + 2 * lane] = tv;
        }
        __syncthreads();

        // prefetch next tile's K rows into cache (gfx1250 global_prefetch_b8)
        if (t + 1 < NTILES) {
            int tn = t + 1;
            int pn;
            if (tn < 3) {
                pn = qbase - WIN + tn * 16 + m;
                pn = pn < 0 ? 0 : (pn > S_LEN - 1 ? S_LEN - 1 : pn);
            } else {
                pn = ((tn - 3) * 16 + m) * KSTRIDE;
            }
            __builtin_prefetch(k + hbase + (long)pn * DH, 0, 3);
        }

        // 16x16 score tile via 16 fp32 WMMAs over the K=64 head dim
        v8f acc = {};
        #pragma unroll
        for (int kc = 0; kc < 16; ++kc) {
            const int d0 = 4 * kc + 2 * hi;   // lanes<16: K=0,1  lanes>=16: K=2,3
            v2f a = *(const v2f*)&q_lds[m * ROWPAD + d0];
            v2f b = *(const v2f*)&k_lds[m * ROWPAD + d0];
            acc = __builtin_amdgcn_wmma_f32_16x16x4_f32(
                false, a, false, b, (short)0, acc, false, false);
        }

        // scale + mask + spill to score LDS.
        // D layout: lane half hi, VGPR r -> (M = r + 8*hi, N = lane&15)
        #pragma unroll
        for (int r = 0; r < 8; ++r) {
            const int row = r + 8 * hi;       // query row in block
            const int n   = qbase + row;      // global query index
            const int col = t * 16 + m;       // score column 0..111
            bool valid;
            if (t < 3) {
                const int p = qbase - WIN + col;               // global key pos
                valid = (col >= row) && (col <= row + 2 * WIN) // window membership
                        && (p >= 0) && (p < S_LEN);            // sequence bounds
            } else {
                valid = (n > WIN);                             // use_str
            }
            sc_lds[row * SCPAD + col] = valid ? acc[r] * 0.125f : -1.0e30f;
        }
        __syncthreads();
    }

    // ================= softmax: one row per lane (lanes 0..15) =================
    if (lane < 16) {
        float mx = -1.0e30f;
        for (int j = 0; j < NCOLS; ++j)
            mx = fmaxf(mx, sc_lds[lane * SCPAD + j]);
        float sum = 0.0f;
        for (int j = 0; j < NCOLS; ++j) {
            float e = __expf(sc_lds[lane * SCPAD + j] - mx);
            sc_lds[lane * SCPAD + j] = e;
            sum += e;
        }
        const float inv = (sum > 0.0f) ? (1.0f / sum) : 0.0f;
        for (int j = 0; j < NCOLS; ++j)
            sc_lds[lane * SCPAD + j] *= inv;
    }
    __syncthreads();

    // ================= PV phase: O = A_loc V_loc + A_str V_str =================
    v8f acc_o[4] = {v8f{}, v8f{}, v8f{}, v8f{}};
    for (int t = 0; t < NTILES; ++t) {
        for (int r = 0; r < 16; ++r) {
            int pos;
            if (t < 3) {
                pos = qbase - WIN + t * 16 + r;
                pos = pos < 0 ? 0 : (pos > S_LEN - 1 ? S_LEN - 1 : pos);
            } else {
                pos = ((t - 3) * 16 + r) * KSTRIDE;
            }
            const float* src = v + hbase + (long)pos * DH + 2 * lane;
            v2f tv; tv.x = src[0]; tv.y = src[1];
            *(v2f*)&v_lds[r * ROWPAD + 2 * lane] = tv;
        }
        __syncthreads();

        #pragma unroll
        for (int kc = 0; kc < 4; ++kc) {
            const int c0 = t * 16 + 4 * kc + 2 * hi;   // prob columns (K of PV)
            const int k0 = 4 * kc + 2 * hi;            // V rows for B operand
            v2f a = *(const v2f*)&sc_lds[m * SCPAD + c0];
            #pragma unroll
            for (int nt = 0; nt < 4; ++nt) {           // output dim tiles
                v2f b;
                b.x = v_lds[(k0 + 0) * ROWPAD + 16 * nt + m];
                b.y = v_lds[(k0 + 1) * ROWPAD + 16 * nt + m];
                acc_o[nt] = __builtin_amdgcn_wmma_f32_16x16x4_f32(
                    false, a, false, b, (short)0, acc_o[nt], false, false);
            }
        }
        __syncthreads();
    }

    // ---- write 16x64 output tile ----
    #pragma unroll
    for (int nt = 0; nt < 4; ++nt) {
        #pragma unroll
        for (int r = 0; r < 8; ++r) {
            const int row = r + 8 * hi;
            out[hbase + (long)(qbase + row) * DH + 16 * nt + m] = acc_o[nt][r];
        }
    }
}

extern "C" void kernel_launch(void* const* d_in, const int* in_sizes, int n_in,
                              void* d_out, int out_size, void* d_ws, size_t ws_size,
                              hipStream_t stream) {
    const float* q = (const float*)d_in[0];
    const float* k = (const float*)d_in[1];
    const float* v = (const float*)d_in[2];
    float* out = (float*)d_out;

    dim3 grid(S_LEN / 16, NHEAD, 1);   // 128 query blocks x 16 heads = 2048 waves
    dim3 block(32, 1, 1);              // one wave32 per workgroup
    strided_attn_wmma_f32<<<grid, block, 0, stream>>>(q, k, v, out);
}